// GradualAACBR_51719996178591
// MI455X (gfx1250) — compile-verified
//
#include <hip/hip_runtime.h>
#include <math.h>

// Problem constants (match reference)
#define N_TRAIN 1024
#define NY      2
#define DIM     128
#define BQ      256
#define KITER   16
#define TT      (N_TRAIN + NY)   // 1026
#define TP      1088             // padded to multiple of 64 (async tile depth), 32, 16
#define SCALEC  0.25f
#define GAMMAC  0.05f
#define TAUC    256.0f           // 2*D
#define KT2     64               // K-depth of LDS-staged B tile in k_iter

typedef __attribute__((ext_vector_type(2))) float v2f;
typedef __attribute__((ext_vector_type(8))) float v8f;

__device__ __forceinline__ float sigmoidf_(float x) {
    return 1.0f / (1.0f + __expf(-x));
}

__device__ __forceinline__ int label_of(int t, const int* __restrict__ ytr) {
    return (t < N_TRAIN) ? ytr[t] : (t - N_TRAIN);   // y_default = arange(Y)
}

// ---------------------------------------------------------------------------
// K1: per-row stats: p = X.w_order, bl = X.w_base, xsq = |X|^2  (one wave/row)
// ---------------------------------------------------------------------------
__global__ void k_row_stats(const float* __restrict__ Xtr, const float* __restrict__ Xdef,
                            const float* __restrict__ wo,  const float* __restrict__ wb,
                            float* __restrict__ p, float* __restrict__ bl,
                            float* __restrict__ xsq) {
    int t    = blockIdx.x * blockDim.y + threadIdx.y;   // blockDim = (32,8)
    int lane = threadIdx.x;
    if (t >= TP) return;
    float sp = 0.f, sb = 0.f, sq = 0.f;
    if (t < TT) {
        const float* x = (t < N_TRAIN) ? (Xtr + (size_t)t * DIM)
                                       : (Xdef + (size_t)(t - N_TRAIN) * DIM);
#pragma unroll
        for (int q = 0; q < 4; ++q) {
            int d = lane + 32 * q;          // coalesced across the wave
            float xv = x[d];
            sp = fmaf(xv, wo[d], sp);
            sb = fmaf(xv, wb[d], sb);
            sq = fmaf(xv, xv, sq);
        }
    }
#pragma unroll
    for (int o = 16; o > 0; o >>= 1) {
        sp += __shfl_xor(sp, o, 32);
        sb += __shfl_xor(sb, o, 32);
        sq += __shfl_xor(sq, o, 32);
    }
    if (lane == 0) { p[t] = sp; bl[t] = sb; xsq[t] = sq; }
}

// ---------------------------------------------------------------------------
// K2: build Es (default rows zeroed) and Wm (same-label, zero diag), padded 0
// ---------------------------------------------------------------------------
__global__ void k_build(const float* __restrict__ p, const int* __restrict__ ytr,
                        float* __restrict__ Es, float* __restrict__ Wm) {
    int j = blockIdx.x * blockDim.x + threadIdx.x;
    int i = blockIdx.y;
    if (j >= TP) return;
    float es = 0.f, wm = 0.f;
    if (i < TT && j < TT) {
        float pi = p[i], pj = p[j];
        float Eij = sigmoidf_(SCALEC * (pi - pj));
        float Eji = sigmoidf_(SCALEC * (pj - pi));
        float e = Eij * (1.0f - Eji);
        if (i >= N_TRAIN) e = 0.0f;                       // defaults don't attack
        bool same = (label_of(i, ytr) == label_of(j, ytr));
        es = e;
        wm = (same && (i != j)) ? e : 0.0f;
    }
    Es[(size_t)i * TP + j] = es;
    Wm[(size_t)i * TP + j] = wm;
}

// ---------------------------------------------------------------------------
// K3: blocked[i,j] = prod_k (1 - Wm[i,k]*Es[k,j]); fuse final A assembly.
// 32x32 tile per 256-thread block, 2x2 register micro-tile, LDS-staged.
// Product accumulation is not bilinear -> VALU kernel by necessity.
// ---------------------------------------------------------------------------
#define TK 32
__global__ void k_blocked(const float* __restrict__ Es, const float* __restrict__ Wm,
                          const float* __restrict__ p,  const int* __restrict__ ytr,
                          float* __restrict__ A) {
    __shared__ float lw[TK][TK + 1];
    __shared__ float le[TK][TK + 1];
    int tid = threadIdx.x;                 // 256
    int i0 = blockIdx.y * TK, j0 = blockIdx.x * TK;
    int ty = tid >> 4, tx = tid & 15;
    float a00 = 1.f, a01 = 1.f, a10 = 1.f, a11 = 1.f;
    for (int kc = 0; kc < TP; kc += TK) {
#pragma unroll
        for (int q = 0; q < 4; ++q) {
            int idx = tid + 256 * q;
            int r = idx >> 5, c = idx & 31;
            lw[r][c] = Wm[(size_t)(i0 + r) * TP + kc + c];
            le[r][c] = Es[(size_t)(kc + r) * TP + j0 + c];
        }
        __syncthreads();
#pragma unroll
        for (int k = 0; k < TK; ++k) {
            float w0 = lw[2 * ty][k],     w1 = lw[2 * ty + 1][k];
            float e0 = le[k][2 * tx],     e1 = le[k][2 * tx + 1];
            a00 *= fmaf(-w0, e0, 1.0f);
            a01 *= fmaf(-w0, e1, 1.0f);
            a10 *= fmaf(-w1, e0, 1.0f);
            a11 *= fmaf(-w1, e1, 1.0f);
        }
        __syncthreads();
    }
    float accs[2][2] = {{a00, a01}, {a10, a11}};
#pragma unroll
    for (int di = 0; di < 2; ++di)
#pragma unroll
        for (int dj = 0; dj < 2; ++dj) {
            int i = i0 + 2 * ty + di, j = j0 + 2 * tx + dj;
            float out = 0.0f;
            if (i < TT && j < TT) {
                bool same = (label_of(i, ytr) == label_of(j, ytr));
                float es = Es[(size_t)i * TP + j];       // already default-masked
                float attacks = same ? 0.0f : es;
                float Eij = sigmoidf_(SCALEC * (p[i] - p[j]));
                float Eji = sigmoidf_(SCALEC * (p[j] - p[i]));
                float sym = (same || i >= N_TRAIN) ? 0.0f : (Eij * Eji);
                out = -(attacks * accs[di][dj] + sym);
            }
            A[(size_t)i * TP + j] = out;
        }
}

// ---------------------------------------------------------------------------
// K4: r[b,t] = sigma(GAMMA*(|q|^2 + |x|^2 - 2 q.x - TAU));  s0[b,t]=sigma(bl[t])
// 16x16 output tile per block, LDS-staged X rows (D=128).
// ---------------------------------------------------------------------------
__global__ void k_r_s0(const float* __restrict__ Xq, const float* __restrict__ Xtr,
                       const float* __restrict__ Xdef, const float* __restrict__ xsq,
                       const float* __restrict__ bl, float* __restrict__ r,
                       float* __restrict__ s0) {
    __shared__ float lq[16][DIM];
    __shared__ float lx[16][DIM];
    int tid = threadIdx.x;                 // 256
    int b0 = blockIdx.y * 16, t0 = blockIdx.x * 16;
#pragma unroll
    for (int q = 0; q < 8; ++q) {
        int idx = tid + 256 * q;
        int row = idx >> 7, col = idx & 127;
        lq[row][col] = Xq[(size_t)(b0 + row) * DIM + col];
        int t = t0 + row;
        float xv = 0.0f;
        if (t < N_TRAIN)      xv = Xtr[(size_t)t * DIM + col];
        else if (t < TT)      xv = Xdef[(size_t)(t - N_TRAIN) * DIM + col];
        lx[row][col] = xv;
    }
    __syncthreads();
    int ty = tid >> 4, tx = tid & 15;      // ty -> query b, tx -> case t
    float dot = 0.f, qsq = 0.f;
#pragma unroll 8
    for (int d = 0; d < DIM; ++d) {
        float qv = lq[ty][d];
        dot = fmaf(qv, lx[tx][d], dot);
        qsq = fmaf(qv, qv, qsq);
    }
    int b = b0 + ty, t = t0 + tx;
    float rv = 0.0f;
    if (t < TT) rv = sigmoidf_(GAMMAC * (qsq + xsq[t] - 2.0f * dot - TAUC));
    r[(size_t)b * TP + t]  = rv;
    s0[(size_t)b * TP + t] = sigmoidf_(bl[t]);          // bl[t>=T] == 0 -> 0.5, harmless
}

// ---------------------------------------------------------------------------
// K5: one fixed-point step  s_next = sigma(bl + s @ A - r)  via f32 WMMA.
// One workgroup (512 threads = 16 waves) per 16-column n-tile; wave w owns
// query-row tile m0 = 16*w (covers all B=256 rows). The shared B operand
// (A[kc:kc+64, n0:n0+16], 4 KB) is staged into LDS ONCE per workgroup with
// gfx1250 async-to-LDS loads (bypasses VGPRs, tracked by ASYNCcnt), cutting
// B-operand L2 traffic 16x vs per-wave loads.
//   A-op (s rows): lane l -> row l&15, K pair 2*(l>>4)   (one 8B load)
//   B-op (LDS):    lane l -> col l&15, K rows {kk+(l>>4), kk+2+(l>>4)}
//   C/D: VGPR v   -> row m0 + v + 8*(l>>4), col n0 + (l&15)
// ---------------------------------------------------------------------------
__global__ void __launch_bounds__(512)
k_iter(const float* __restrict__ S, const float* __restrict__ Amat,
       const float* __restrict__ bl, const float* __restrict__ r,
       float* __restrict__ Snext) {
    __shared__ __align__(16) float bt[KT2 * 16];   // 64 x 16 B tile (4 KB)
    int tid   = threadIdx.x;                // 0..511
    int lane  = tid & 31;
    int wv    = tid >> 5;                   // 0..15 -> m-tile
    int n0    = blockIdx.x * 16;            // target-col tile
    int m0    = wv * 16;                    // query-row tile
    int mrow  = lane & 15;
    int khalf = lane >> 4;                  // 0 or 1
    v8f acc = {};
    const float* srow = S + (size_t)(m0 + mrow) * TP;

    // staging map: thread -> one b64 (2 floats) of the 64x16 tile
    int sidx  = tid * 2;                    // 0..1022
    int srk   = sidx >> 4;                  // local k row 0..63
    int scol  = sidx & 15;                  // col (even)
    unsigned ldsoff = (unsigned)(size_t)(bt + sidx);   // LDS byte offset

    for (int kc = 0; kc < TP; kc += KT2) {
        const float* gsrc = Amat + (size_t)(kc + srk) * TP + n0 + scol;
        // async copy global -> LDS (no VGPR data path), ASYNCcnt tracked
        asm volatile("global_load_async_to_lds_b64 %0, %1, off"
                     :: "v"(ldsoff), "v"(gsrc)
                     : "memory");
        asm volatile("s_wait_asynccnt 0x0" ::: "memory");  // own writes landed
        __syncthreads();                                    // all waves' writes visible
#pragma unroll
        for (int kk = 0; kk < KT2; kk += 4) {
            v2f a = *(const v2f*)(srow + kc + kk + 2 * khalf);
            v2f b;
            b.x = bt[(kk + khalf) * 16 + mrow];
            b.y = bt[(kk + khalf + 2) * 16 + mrow];
            acc = __builtin_amdgcn_wmma_f32_16x16x4_f32(
                /*neg_a=*/false, a, /*neg_b=*/false, b,
                /*c_mod=*/(short)0, acc, /*reuse_a=*/false, /*reuse_b=*/false);
        }
        __syncthreads();                    // tile fully consumed before overwrite
    }
    int col = n0 + mrow;
    float blc = bl[col];
#pragma unroll
    for (int v = 0; v < 8; ++v) {
        int row = m0 + v + khalf * 8;
        float x = blc + acc[v] - r[(size_t)row * TP + col];
        Snext[(size_t)row * TP + col] = sigmoidf_(x);
    }
}

// ---------------------------------------------------------------------------
// K6: gather the two default-argument strengths per query.
// ---------------------------------------------------------------------------
__global__ void k_out(const float* __restrict__ S, float* __restrict__ out) {
    int idx = blockIdx.x * blockDim.x + threadIdx.x;
    if (idx >= BQ * NY) return;
    int b = idx / NY, y = idx % NY;
    out[idx] = S[(size_t)b * TP + N_TRAIN + y];
}

// ---------------------------------------------------------------------------
// Host-side launcher.  Workspace layout (floats):
//   p[TP] | bl[TP] | xsq[TP] | Es[TP*TP] | Wm[TP*TP] | A[TP*TP]
//   | r[B*TP] | s0[B*TP] | s1[B*TP]          (~18 MB total)
// ---------------------------------------------------------------------------
extern "C" void kernel_launch(void* const* d_in, const int* in_sizes, int n_in,
                              void* d_out, int out_size, void* d_ws, size_t ws_size,
                              hipStream_t stream) {
    const float* Xtr  = (const float*)d_in[0];
    const float* Xdef = (const float*)d_in[1];
    const float* Xq   = (const float*)d_in[2];
    const float* wo   = (const float*)d_in[3];
    const float* wb   = (const float*)d_in[4];
    const int*   ytr  = (const int*)d_in[5];
    // d_in[6] (y_default_labels) == arange(Y), handled analytically.

    float* ws  = (float*)d_ws;
    float* p   = ws;
    float* bl  = ws + TP;
    float* xsq = ws + 2 * (size_t)TP;
    float* Es  = ws + 3 * (size_t)TP;
    float* Wm  = Es + (size_t)TP * TP;
    float* A   = Wm + (size_t)TP * TP;
    float* r   = A  + (size_t)TP * TP;
    float* s0  = r  + (size_t)BQ * TP;
    float* s1  = s0 + (size_t)BQ * TP;

    k_row_stats<<<dim3(TP / 8), dim3(32, 8), 0, stream>>>(Xtr, Xdef, wo, wb, p, bl, xsq);
    k_build    <<<dim3((TP + 255) / 256, TP), 256, 0, stream>>>(p, ytr, Es, Wm);
    k_blocked  <<<dim3(TP / TK, TP / TK), 256, 0, stream>>>(Es, Wm, p, ytr, A);
    k_r_s0     <<<dim3(TP / 16, BQ / 16), 256, 0, stream>>>(Xq, Xtr, Xdef, xsq, bl, r, s0);

    float* cur = s0;
    float* nxt = s1;
    for (int it = 0; it < KITER; ++it) {
        k_iter<<<dim3(TP / 16), 512, 0, stream>>>(cur, A, bl, r, nxt);
        float* tmp = cur; cur = nxt; nxt = tmp;
    }
    k_out<<<dim3((BQ * NY + 255) / 256), 256, 0, stream>>>(cur, (float*)d_out);
}